// FlowfromDepthLayer_32538672235129
// MI455X (gfx1250) — compile-verified
//
#include <hip/hip_runtime.h>
#include <hip/hip_bf16.h>
#include <stdint.h>

// ---------------------------------------------------------------------------
// FlowFromDepth: B=32, H=512, W=640. Bandwidth-bound streaming kernel.
// 16 B/pixel * 10.49 Mpix ~= 168 MB -> ~7.2 us at 23.3 TB/s HBM; the whole
// working set fits in the 192 MB global L2 for graph replays.
// CDNA5 paths: global_load_async_to_lds_b128 (ASYNCcnt) staging +
// s_wait_asynccnt + ds_load_b128 readback; b128 stores to both flow planes.
// WMMA intentionally unused (AI ~ 2 FLOP/byte -> memory roofline).
// 8 px/thread to amortize the uniform per-thread setup (matrix regs, row
// math, counter wait) over twice the traffic.
// ---------------------------------------------------------------------------

namespace {
constexpr int kH = 512;
constexpr int kW = 640;
constexpr int kHW = kH * kW;
constexpr int kThreads = 256;        // 8 wave32s per block
constexpr int kPixPerThread = 8;     // 2 x b128 per array per thread
constexpr int kPixPerBlock = kThreads * kPixPerThread;  // 2048; 327680/2048=160
}

__device__ __forceinline__ void inv3x3(const float a[9], float o[9]) {
  const float c00 = a[4] * a[8] - a[5] * a[7];
  const float c10 = a[5] * a[6] - a[3] * a[8];
  const float c20 = a[3] * a[7] - a[4] * a[6];
  const float det = a[0] * c00 + a[1] * c10 + a[2] * c20;
  const float id = 1.0f / det;
  o[0] = c00 * id;
  o[1] = (a[2] * a[7] - a[1] * a[8]) * id;
  o[2] = (a[1] * a[5] - a[2] * a[4]) * id;
  o[3] = c10 * id;
  o[4] = (a[0] * a[8] - a[2] * a[6]) * id;
  o[5] = (a[2] * a[3] - a[0] * a[5]) * id;
  o[6] = c20 * id;
  o[7] = (a[1] * a[6] - a[0] * a[7]) * id;
  o[8] = (a[0] * a[4] - a[1] * a[3]) * id;
}

// Per-batch setup: M = (K R^T) K^-1 (9 floats), Wv = (K R^T)(-t) (3 floats).
// mw layout per batch: [M00..M22, Wx, Wy, Wz] (12 floats)
__global__ void flow_precompute(const float* __restrict__ trans,
                                const float* __restrict__ rot,
                                const float* __restrict__ intr,
                                float* __restrict__ mw, int nB) {
  const int b = blockIdx.x * blockDim.x + threadIdx.x;
  if (b >= nB) return;
  float K[9], R[9], t3[3];
#pragma unroll
  for (int i = 0; i < 9; ++i) {
    K[i] = intr[b * 9 + i];
    R[i] = rot[b * 9 + i];
  }
  t3[0] = trans[b * 3 + 0];
  t3[1] = trans[b * 3 + 1];
  t3[2] = trans[b * 3 + 2];

  float Ki[9];
  inv3x3(K, Ki);

  float T[9];  // K @ R^T
#pragma unroll
  for (int i = 0; i < 3; ++i)
#pragma unroll
    for (int j = 0; j < 3; ++j)
      T[i * 3 + j] = K[i * 3 + 0] * R[j * 3 + 0] +
                     K[i * 3 + 1] * R[j * 3 + 1] +
                     K[i * 3 + 2] * R[j * 3 + 2];

  float* o = mw + b * 12;
#pragma unroll
  for (int i = 0; i < 3; ++i) {
#pragma unroll
    for (int j = 0; j < 3; ++j)
      o[i * 3 + j] = T[i * 3 + 0] * Ki[0 * 3 + j] +
                     T[i * 3 + 1] * Ki[1 * 3 + j] +
                     T[i * 3 + 2] * Ki[2 * 3 + j];
    o[9 + i] = -(T[i * 3 + 0] * t3[0] + T[i * 3 + 1] * t3[1] +
                 T[i * 3 + 2] * t3[2]);
  }
}

__global__ __launch_bounds__(kThreads) void flow_main(
    const float* __restrict__ depth, const float* __restrict__ mask,
    const float* __restrict__ mw, float* __restrict__ out) {
  __shared__ __align__(16) float sD[kThreads * kPixPerThread];
  __shared__ __align__(16) float sM[kThreads * kPixPerThread];

  const int b = blockIdx.y;
  const int tid = threadIdx.x;
  const int p = blockIdx.x * kPixPerBlock + tid * kPixPerThread;

  // ---- stage depth + mask (2 x b128 each per lane) via CDNA5 async path ----
  const float* gD = depth + (size_t)b * kHW + p;
  const float* gM = mask + (size_t)b * kHW + p;
  float* sDp = &sD[tid * kPixPerThread];
  float* sMp = &sM[tid * kPixPerThread];
#pragma unroll
  for (int h = 0; h < kPixPerThread / 4; ++h) {
    const uint32_t lD = (uint32_t)(uintptr_t)(sDp + h * 4);
    const uint32_t lM = (uint32_t)(uintptr_t)(sMp + h * 4);
    asm volatile("global_load_async_to_lds_b128 %0, %1, off"
                 :
                 : "v"(lD), "v"((uint64_t)(uintptr_t)(gD + h * 4))
                 : "memory");
    asm volatile("global_load_async_to_lds_b128 %0, %1, off"
                 :
                 : "v"(lM), "v"((uint64_t)(uintptr_t)(gM + h * 4))
                 : "memory");
  }

  // ---- overlap: per-batch M / Wv (uniform across block) + grid math ----
  const float* c = mw + b * 12;
  const float M00 = c[0], M01 = c[1], M02 = c[2];
  const float M10 = c[3], M11 = c[4], M12 = c[5];
  const float M20 = c[6], M21 = c[7], M22 = c[8];
  const float Wx = c[9], Wy = c[10], Wz = c[11];

  const int row = p / kW;  // 8 consecutive pixels share a row (640 % 8 == 0)
  const float y = (float)row;
  const float x0 = (float)(p - row * kW);

  // inter(x0) = M @ [x0, y, 1]; inter(x0+j) = inter(x0) + j*M[:,0]
  const float ix0 = fmaf(M00, x0, fmaf(M01, y, M02));
  const float iy0 = fmaf(M10, x0, fmaf(M11, y, M12));
  const float iz0 = fmaf(M20, x0, fmaf(M21, y, M22));

  // ---- wait on our own async transfers (per-wave ASYNCcnt), read back ----
  asm volatile("s_wait_asynccnt 0x0" ::: "memory");

  float dd[kPixPerThread], mm[kPixPerThread];
#pragma unroll
  for (int h = 0; h < kPixPerThread / 4; ++h) {
    const float4 d4 = *(const float4*)(sDp + h * 4);
    const float4 m4 = *(const float4*)(sMp + h * 4);
    dd[h * 4 + 0] = d4.x; dd[h * 4 + 1] = d4.y;
    dd[h * 4 + 2] = d4.z; dd[h * 4 + 3] = d4.w;
    mm[h * 4 + 0] = m4.x; mm[h * 4 + 1] = m4.y;
    mm[h * 4 + 2] = m4.z; mm[h * 4 + 3] = m4.w;
  }

  float u[kPixPerThread], v[kPixPerThread];
#pragma unroll
  for (int j = 0; j < kPixPerThread; ++j) {
    const float jx = fmaf((float)j, M00, ix0);
    const float jy = fmaf((float)j, M10, iy0);
    const float jz = fmaf((float)j, M20, iz0);
    const float nx = fmaf(dd[j], jx, Wx);
    const float ny = fmaf(dd[j], jy, Wy);
    const float nz = fmaf(dd[j], jz, Wz);
    // depth2 = 1e30*(1-m) + m*nz
    const float depth2 = fmaf(1e30f, 1.0f - mm[j], mm[j] * nz);
    const float inv = __builtin_amdgcn_rcpf(depth2);  // v_rcp_f32, ~1 ulp
    u[j] = fmaf(nx, inv, -(x0 + (float)j)) * (1.0f / (float)kW);
    v[j] = fmaf(ny, inv, -y) * (1.0f / (float)kH);
  }

  // ---- b128 stores to the two contiguous flow planes of [B,2,H,W] ----
  float* ou = out + (size_t)b * 2 * kHW + p;
  float* ov = ou + kHW;
#pragma unroll
  for (int h = 0; h < kPixPerThread / 4; ++h) {
    *(float4*)(ou + h * 4) =
        make_float4(u[h * 4 + 0], u[h * 4 + 1], u[h * 4 + 2], u[h * 4 + 3]);
    *(float4*)(ov + h * 4) =
        make_float4(v[h * 4 + 0], v[h * 4 + 1], v[h * 4 + 2], v[h * 4 + 3]);
  }
}

extern "C" void kernel_launch(void* const* d_in, const int* in_sizes, int n_in,
                              void* d_out, int out_size, void* d_ws,
                              size_t ws_size, hipStream_t stream) {
  const float* depth = (const float*)d_in[0];  // [B,1,H,W]
  const float* mask = (const float*)d_in[1];   // [B,1,H,W]
  const float* trans = (const float*)d_in[2];  // [B,3,1]
  const float* rot = (const float*)d_in[3];    // [B,3,3]
  const float* intr = (const float*)d_in[4];   // [B,3,3]
  float* out = (float*)d_out;                  // [B,2,H,W]
  float* mw = (float*)d_ws;                    // B*12 floats (1.5 KB)

  const int B = in_sizes[2] / 3;

  flow_precompute<<<1, (B > 0 ? B : 1), 0, stream>>>(trans, rot, intr, mw, B);

  dim3 grid(kHW / kPixPerBlock, B);  // 160 x 32 blocks, exact cover
  flow_main<<<grid, kThreads, 0, stream>>>(depth, mask, mw, out);
}